// NormalizedCrossCorrelationLoss_62173946577083
// MI455X (gfx1250) — compile-verified
//
#include <hip/hip_runtime.h>
#include <stdint.h>

// Geometry
#define DIN 160
#define HIN 192
#define WIN 160
#define DOUT 154
#define HOUT 186
#define WOUT 154
#define TIN 22            // 16 + 6 halo
#define ZCHUNK 4
#define CH ((DOUT + ZCHUNK - 1) / ZCHUNK)   // 39
#define NBLK (10 * 12 * 2 * ZCHUNK)         // 960
#define NWK 343.0f

typedef float v2f __attribute__((ext_vector_type(2)));
typedef float v8f __attribute__((ext_vector_type(8)));

// ---- CDNA5 async global->LDS copy (ASYNCcnt path) ----
__device__ __forceinline__ void async_ld_b32(uint32_t lds_off, const float* gptr) {
    asm volatile("global_load_async_to_lds_b32 %0, %1, off"
                 :: "v"(lds_off), "v"((uint64_t)(uintptr_t)gptr) : "memory");
}
__device__ __forceinline__ uint32_t lds_off(const void* p) {
    return (uint32_t)(uintptr_t)p;   // low 32 bits of flat addr == LDS byte address
}

// Issue one 22x22 plane of both inputs into ring slot (4 async b32 per thread).
__device__ __forceinline__ void load_plane(float* dX, float* dY,
                                           const float* Xb, const float* Yb,
                                           int z, int x0, int y0, int t) {
#pragma unroll
    for (int rep = 0; rep < 2; ++rep) {
        int p = t + rep * 256;
        p = (p < TIN * TIN) ? p : t;            // duplicate tail loads (harmless)
        int r = p / TIN, c = p - r * TIN;
        int yi = y0 + r; yi = (yi > HIN - 1) ? HIN - 1 : yi;   // clamp halo
        int xi = x0 + c; xi = (xi > WIN - 1) ? WIN - 1 : xi;
        size_t g = ((size_t)z * HIN + yi) * (size_t)WIN + (size_t)xi;
        async_ld_b32(lds_off(dX + r * 24 + c), Xb + g);
        async_ld_b32(lds_off(dY + r * 24 + c), Yb + g);
    }
}

__global__ __launch_bounds__(256)
void ncc_main(const float* __restrict__ X, const float* __restrict__ Y,
              float* __restrict__ ws) {
    __shared__ float smRX[8][22 * 24];    // z-ring of raw x tiles
    __shared__ float smRY[8][22 * 24];    // z-ring of raw y tiles
    __shared__ float smQT[5][24 * 33];    // z-summed quantities, transposed [xx][r]
    __shared__ float smS1[5][24 * 17];    // after x-pass: [yy][xo]
    __shared__ float smOut[5][256];       // final 16x16 box sums per quantity
    __shared__ float smRed[5][8];

    const int t    = threadIdx.x;
    const int lane = t & 31;
    const int wv   = t >> 5;
    const int x0 = blockIdx.x * 16, y0 = blockIdx.y * 16;
    const int b  = blockIdx.z / ZCHUNK, zc = blockIdx.z % ZCHUNK;
    const int z0 = zc * CH;
    const int z1 = (z0 + CH < DOUT) ? (z0 + CH) : DOUT;
    const float* Xb = X + (size_t)b * DIN * HIN * WIN;
    const float* Yb = Y + (size_t)b * DIN * HIN * WIN;

    // zero pad regions once (writes each iteration only touch r<22,c<22 / rows<22)
    for (int i = t; i < 5 * 24 * 33; i += 256) (&smQT[0][0])[i] = 0.0f;
    for (int i = t; i < 5 * 24 * 17; i += 256) (&smS1[0][0])[i] = 0.0f;

    // prologue: planes z0..z0+6 into ring (slot = z&7)
#pragma unroll 1
    for (int z = z0; z < z0 + 7; ++z)
        load_plane(smRX[z & 7], smRY[z & 7], Xb, Yb, z, x0, y0, t);
    asm volatile("s_wait_asynccnt 0x0" ::: "memory");
    __syncthreads();

    float minv = __builtin_inff();
    float sv = 0.0f, sn = 0.0f, snv = 0.0f, cnt = 0.0f;
    const int m     = lane & 15;                // WMMA row / column helper
    const int khalf = (lane < 16) ? 0 : 2;
    const int hi    = (lane >= 16) ? 8 : 0;

#pragma unroll 1
    for (int zo = z0; zo < z1; ++zo) {
        // software-pipelined ring refill: plane zo+7 -> slot (zo+7)&7
        const int zin = zo + 7;
        if (zin <= z1 + 5) {
            load_plane(smRX[zin & 7], smRY[zin & 7], Xb, Yb, zin, x0, y0, t);
            asm volatile("s_wait_asynccnt 0x4" ::: "memory"); // prev plane landed
        } else {
            asm volatile("s_wait_asynccnt 0x0" ::: "memory");
        }
        __syncthreads();   // A: planes zo..zo+6 resident

        // z-direction 7-sum of the 5 quantities (per-element products first)
#pragma unroll
        for (int rep = 0; rep < 2; ++rep) {
            int p = t + rep * 256;
            if (p < TIN * TIN) {
                int r = p / TIN, c = p - r * TIN;
                float ax = 0.f, ay = 0.f, axx = 0.f, ayy = 0.f, axy = 0.f;
#pragma unroll
                for (int dz = 0; dz < 7; ++dz) {
                    int sl = (zo + dz) & 7;
                    float xv = smRX[sl][r * 24 + c];
                    float yv = smRY[sl][r * 24 + c];
                    ax += xv; ay += yv;
                    axx += xv * xv; ayy += yv * yv; axy += xv * yv;
                }
                int o = c * 33 + r;           // transposed: row = xx, col = r
                smQT[0][o] = ax;  smQT[1][o] = ay;
                smQT[2][o] = axx; smQT[3][o] = ayy; smQT[4][o] = axy;
            }
        }
        __syncthreads();   // B: smQT complete

        // Banded-matmul box sums on the matrix pipe: waves 0..4, one quantity each
        if (wv < 5) {
            const float* QT = smQT[wv];
            float* S1w = smS1[wv];
            // x-pass: S1[xo][r] = sum_xx Band[xo][xx] * QT[xx][r], two 16-col tiles
#pragma unroll
            for (int h = 0; h < 2; ++h) {
                v8f acc = {0.f, 0.f, 0.f, 0.f, 0.f, 0.f, 0.f, 0.f};
#pragma unroll
                for (int c6 = 0; c6 < 6; ++c6) {
                    int k = 4 * c6 + khalf;
                    v2f a, bf;
                    a[0] = (k     >= m && k     <= m + 6) ? 1.0f : 0.0f;
                    a[1] = (k + 1 >= m && k + 1 <= m + 6) ? 1.0f : 0.0f;
                    int col = h * 16 + m;
                    bf[0] = QT[k * 33 + col];
                    bf[1] = QT[(k + 1) * 33 + col];
                    acc = __builtin_amdgcn_wmma_f32_16x16x4_f32(
                        false, a, false, bf, (short)0, acc, false, false);
                }
                int r = h * 16 + m;           // D column index = y-row r
                if (r < 22) {
#pragma unroll
                    for (int j = 0; j < 8; ++j)
                        S1w[r * 17 + (j + hi)] = acc[j];   // store as [yy][xo]
                }
            }
            // y-pass: Out[yo][xo] = sum_yy Band[yo][yy] * S1[yy][xo]
            {
                v8f acc = {0.f, 0.f, 0.f, 0.f, 0.f, 0.f, 0.f, 0.f};
#pragma unroll
                for (int c6 = 0; c6 < 6; ++c6) {
                    int k = 4 * c6 + khalf;
                    v2f a, bf;
                    a[0] = (k     >= m && k     <= m + 6) ? 1.0f : 0.0f;
                    a[1] = (k + 1 >= m && k + 1 <= m + 6) ? 1.0f : 0.0f;
                    bf[0] = S1w[k * 17 + m];
                    bf[1] = S1w[(k + 1) * 17 + m];
                    acc = __builtin_amdgcn_wmma_f32_16x16x4_f32(
                        false, a, false, bf, (short)0, acc, false, false);
                }
#pragma unroll
                for (int j = 0; j < 8; ++j)
                    smOut[wv][(j + hi) * 16 + m] = acc[j];
            }
        }
        __syncthreads();   // C: smOut complete

        // Fused NCC math + masked accumulation (one voxel per thread)
        {
            int yo = t >> 4, xo = t & 15;
            float sx  = smOut[0][t], sy  = smOut[1][t];
            float sxx = smOut[2][t], syy = smOut[3][t], sxy = smOut[4][t];
            float mx = sx * (1.0f / NWK), my = sy * (1.0f / NWK);
            float num = sxy + mx * my * NWK - mx * sy - my * sx;
            float d0  = mx * mx * NWK + sxx - 2.0f * mx * sx;
            float d1  = my * my * NWK + syy - 2.0f * my * sy;
            float den = d0 * d1;
            bool valid = den > 1e-5f;
            float ncc = valid ? (num / sqrtf(den)) : 0.0f;
            bool inv = (ncc < -1.0f - 1e-5f) || (ncc > 1.0f + 1e-5f);
            if (inv) ncc = 0.0f;
            bool av = valid && !inv;
            float v = 0.5f * (d0 + d1);
            bool inr = (x0 + xo < WOUT) && (y0 + yo < HOUT);
            if (av && inr) {
                minv = fminf(minv, v);
                sv += v; sn += ncc; snv += ncc * v; cnt += 1.0f;
            }
        }
        __syncthreads();   // D: smOut reads done before next overwrite
    }

    // block reduction -> deterministic per-block partials
#pragma unroll
    for (int off = 16; off > 0; off >>= 1) {
        minv = fminf(minv, __shfl_down(minv, off, 32));
        sv  += __shfl_down(sv,  off, 32);
        sn  += __shfl_down(sn,  off, 32);
        snv += __shfl_down(snv, off, 32);
        cnt += __shfl_down(cnt, off, 32);
    }
    if (lane == 0) {
        smRed[0][wv] = minv; smRed[1][wv] = sv; smRed[2][wv] = sn;
        smRed[3][wv] = snv;  smRed[4][wv] = cnt;
    }
    __syncthreads();
    if (t == 0) {
        float a0 = smRed[0][0], a1 = smRed[1][0], a2 = smRed[2][0],
              a3 = smRed[3][0], a4 = smRed[4][0];
        for (int w = 1; w < 8; ++w) {
            a0 = fminf(a0, smRed[0][w]); a1 += smRed[1][w]; a2 += smRed[2][w];
            a3 += smRed[3][w]; a4 += smRed[4][w];
        }
        int bid = blockIdx.x + gridDim.x * (blockIdx.y + gridDim.y * blockIdx.z);
        float* e = ws + (size_t)bid * 5;
        e[0] = a0; e[1] = a1; e[2] = a2; e[3] = a3; e[4] = a4;
    }
}

// Deterministic final reduction: vmin + sums -> scalar loss.
// Note: the reference's vmax and HIGH/(vmax-vmin+eps) scale cancel in
// (sum ncc*w)/(sum w); only vmin survives:
//   loss = 1 - (sum ncc*(v-vmin)) / (sum (v-vmin))
__global__ __launch_bounds__(256)
void ncc_finalize(const float* __restrict__ ws, float* __restrict__ out) {
    __shared__ float red[5][8];
    float mv = __builtin_inff(), sv = 0.f, sn = 0.f, snv = 0.f, cnt = 0.f;
    for (int i = threadIdx.x; i < NBLK; i += 256) {
        const float* e = ws + (size_t)i * 5;
        mv = fminf(mv, e[0]); sv += e[1]; sn += e[2]; snv += e[3]; cnt += e[4];
    }
#pragma unroll
    for (int off = 16; off > 0; off >>= 1) {
        mv = fminf(mv, __shfl_down(mv, off, 32));
        sv  += __shfl_down(sv,  off, 32);
        sn  += __shfl_down(sn,  off, 32);
        snv += __shfl_down(snv, off, 32);
        cnt += __shfl_down(cnt, off, 32);
    }
    int wv = threadIdx.x >> 5, ln = threadIdx.x & 31;
    if (ln == 0) { red[0][wv] = mv; red[1][wv] = sv; red[2][wv] = sn;
                   red[3][wv] = snv; red[4][wv] = cnt; }
    __syncthreads();
    if (threadIdx.x == 0) {
        mv = red[0][0]; sv = red[1][0]; sn = red[2][0];
        snv = red[3][0]; cnt = red[4][0];
        for (int w = 1; w < 8; ++w) {
            mv = fminf(mv, red[0][w]); sv += red[1][w]; sn += red[2][w];
            snv += red[3][w]; cnt += red[4][w];
        }
        out[0] = 1.0f - (snv - mv * sn) / (sv - mv * cnt);
    }
}

extern "C" void kernel_launch(void* const* d_in, const int* in_sizes, int n_in,
                              void* d_out, int out_size, void* d_ws, size_t ws_size,
                              hipStream_t stream) {
    (void)in_sizes; (void)n_in; (void)out_size; (void)ws_size;
    const float* X = (const float*)d_in[0];   // y_pred
    const float* Y = (const float*)d_in[1];   // y_true
    // d_in[2] is the ones kernel (sum = 343), folded into constants.
    float* ws = (float*)d_ws;                  // NBLK * 5 floats of partials
    dim3 grid((WOUT + 15) / 16, (HOUT + 15) / 16, 2 * ZCHUNK);
    ncc_main<<<grid, 256, 0, stream>>>(X, Y, ws);
    ncc_finalize<<<1, 256, 0, stream>>>(ws, (float*)d_out);
}